// CovarianceLayer_11081015623766
// MI455X (gfx1250) — compile-verified
//
#include <hip/hip_runtime.h>
#include <math.h>

// ---------------------------------------------------------------------------
// CovarianceLayer: per-(b,c) circular autocorrelation via DFT-as-matmul.
//   d  = x - mean(x)
//   Yr = d@C ; Yi = -d@S                  (row DFT, C/S symmetric 256x256)
//   Zr = C@Yr + S@Yi ; Zi = C@Yi - S@Yr   (col DFT)
//   P  = Zr^2 + Zi^2
//   out = (C@(P@C) - S@(P@S)) / (HW)^2    (real inverse transform)
// All matmuls run on V_WMMA_F32_16X16X4_F32 (exact f32).
// B tiles are stored K-pair-interleaved in LDS so every B fragment is a
// single contiguous ds_load_b64 (no VGPR packing movs before v_wmma).
// ---------------------------------------------------------------------------

typedef __attribute__((ext_vector_type(2))) float v2f;
typedef __attribute__((ext_vector_type(8))) float v8f;

#define NN    256          // image side
#define NIMG  512          // 64 batches * 8 channels
#define IMSZ  65536        // 256*256
#define LDA   36           // LDS row stride for 64x32 A chunks (conflict-free)
#define LDBP  160          // LDS pair stride for B chunks (160 mod 64 == 32)
#define SCALE 2.3283064365386963e-10f   // 1/(65536^2)

__device__ __forceinline__ v8f vzero8() {
  v8f v = {0.f,0.f,0.f,0.f,0.f,0.f,0.f,0.f};
  return v;
}

__device__ __forceinline__ v8f wmma_f32(v2f a, v2f b, v8f c) {
  // (neg_a, A, neg_b, B, c_mod, C, reuse_a, reuse_b); f32 has NEG=CNeg only.
  return __builtin_amdgcn_wmma_f32_16x16x4_f32(false, a, false, b, (short)0, c,
                                               false, false);
}

// Load a 64-row x 32-col f32 tile (global row stride gs) into LDS, stride LDA.
__device__ __forceinline__ void load_tile_A(float* dst, const float* __restrict__ src,
                                            int gs) {
  int t = threadIdx.x;
#pragma unroll
  for (int i = 0; i < 4; ++i) {
    int f   = t + 128 * i;        // float4 index, 512 total
    int row = f >> 3;             // 8 float4 per row
    int c4  = (f & 7) << 2;
    const float4 v = *(const float4*)(src + (size_t)row * gs + c4);
    float* d = dst + row * LDA + c4;
    d[0] = v.x; d[1] = v.y; d[2] = v.z; d[3] = v.w;
  }
}

// Load a 32-row x 64-col f32 tile into LDS, K-pair interleaved:
//   element (row, col) goes to pair row row/2 (stride LDBP), word 2*col, plus 1 if row is odd
// so lane fragments {B[k,n], B[k+1,n]} are one aligned b64 word pair.
__device__ __forceinline__ void load_tile_B(float* dst, const float* __restrict__ src,
                                            int gs) {
  int t = threadIdx.x;
#pragma unroll
  for (int i = 0; i < 4; ++i) {
    int f   = t + 128 * i;        // float4 index, 512 total
    int row = f >> 4;             // 16 float4 per row
    int c4  = (f & 15) << 2;
    const float4 v = *(const float4*)(src + (size_t)row * gs + c4);
    float* d = dst + (row >> 1) * LDBP + (c4 << 1) + (row & 1);
    d[0] = v.x; d[2] = v.y; d[4] = v.z; d[6] = v.w;
  }
}

// A fragment: 16x4, per ISA layout (lanes 0-15: K0,K1; lanes 16-31: K2,K3).
__device__ __forceinline__ v2f fragA(const float* Aw, int ks) {
  int lane = threadIdx.x & 31;
  int m = lane & 15, hi = lane >> 4;
  return *(const v2f*)(Aw + m * LDA + (ks << 2) + (hi << 1));
}

// B fragment from pair-interleaved LDS: single contiguous b64 per lane.
__device__ __forceinline__ v2f fragB(const float* Bs, int ks, int sc) {
  int lane = threadIdx.x & 31;
  int n = lane & 15, hi = lane >> 4;
  int p = (ks << 1) + hi;       // K-pair index: rows {4ks+2hi, 4ks+2hi+1}
  return *(const v2f*)(Bs + p * LDBP + ((sc + n) << 1));
}

// ------------------------- twiddle init -----------------------------------
__global__ __launch_bounds__(256) void ck_twiddles(float* __restrict__ Cm,
                                                   float* __restrict__ Sm) {
  int k = blockIdx.x, n = threadIdx.x;
  int prod = (k * n) & 255;
  float ang = (float)prod * (6.283185307179586f / 256.0f);
  Cm[k * NN + n] = cosf(ang);
  Sm[k * NN + n] = sinf(ang);
}

// --------------------- mean (deterministic, 2 stage) ----------------------
// Block = (b, 32-row slab). Writes partial[(b*8+slab)*8 + c].
__global__ __launch_bounds__(256) void ck_partial(const float* __restrict__ in,
                                                  float* __restrict__ partial) {
  int b  = blockIdx.x >> 3;
  int sl = blockIdx.x & 7;
  const float* p = in + ((size_t)b * 8 + sl) * 65536;  // 32*256*8 floats per slab
  int t = threadIdx.x;
  float acc = 0.f;
#pragma unroll 8
  for (int i = 0; i < 256; ++i) acc += p[t + 256 * i];
  __shared__ float red[256];
  red[t] = acc;
  __syncthreads();
  if (t < 8) {                       // channel t (stride-256 reads fix c = t&7)
    float s = 0.f;
    for (int j = 0; j < 32; ++j) s += red[t + 8 * j];
    partial[(b * 8 + sl) * 8 + t] = s;
  }
}

// Subtract mean and transpose NHWC -> [img][y][x].
__global__ __launch_bounds__(256) void ck_center(const float* __restrict__ in,
                                                 const float* __restrict__ partial,
                                                 float* __restrict__ d) {
  int b  = blockIdx.x >> 3;
  int sl = blockIdx.x & 7;
  int ys = sl << 5;
  __shared__ float mean[8];
  int t = threadIdx.x;
  if (t < 8) {
    float s = 0.f;
    for (int j = 0; j < 8; ++j) s += partial[(b * 8 + j) * 8 + t];  // fixed order
    mean[t] = s * (1.0f / 65536.0f);
  }
  __syncthreads();
  const float* p = in + ((size_t)b * 8 + sl) * 65536;
  int c = t & 7;                     // constant per thread
  float mu = mean[c];
#pragma unroll 4
  for (int i = 0; i < 256; ++i) {
    int f = t + 256 * i;             // flat within slab
    int x = (f >> 3) & 255;
    int y = ys + (f >> 11);
    d[((size_t)(b * 8 + c)) * IMSZ + (size_t)y * NN + x] = p[f] - mu;
  }
}

// ------------------- pass A: outR = A@C ; outI = sI*(A@S) ------------------
__global__ __launch_bounds__(128) void ck_rowdft(const float* __restrict__ Ain,
                                                 const float* __restrict__ Cm,
                                                 const float* __restrict__ Sm,
                                                 float* __restrict__ outR,
                                                 float* __restrict__ outI,
                                                 float sI) {
  __shared__ float As[64 * LDA];
  __shared__ float Bc[16 * LDBP];
  __shared__ float Bs[16 * LDBP];
  int img = blockIdx.x >> 4;
  int tl  = blockIdx.x & 15;
  int r0 = (tl >> 2) << 6, c0 = (tl & 3) << 6;
  const float* Aimg = Ain + (size_t)img * IMSZ;
  int w = threadIdx.x >> 5;

  v8f aR[4], aI[4];
#pragma unroll
  for (int s = 0; s < 4; ++s) { aR[s] = vzero8(); aI[s] = vzero8(); }

  for (int kc = 0; kc < 8; ++kc) {
    __syncthreads();
    load_tile_A(As, Aimg + (size_t)r0 * NN + kc * 32, NN);
    load_tile_B(Bc, Cm + (size_t)(kc * 32) * NN + c0, NN);
    load_tile_B(Bs, Sm + (size_t)(kc * 32) * NN + c0, NN);
    __syncthreads();
    const float* Aw = As + (w << 4) * LDA;
#pragma unroll
    for (int ks = 0; ks < 8; ++ks) {
      v2f a = fragA(Aw, ks);
#pragma unroll
      for (int s = 0; s < 4; ++s) {
        v2f bR = fragB(Bc, ks, s << 4);
        v2f bI = fragB(Bs, ks, s << 4);
        aR[s] = wmma_f32(a, bR, aR[s]);
        aI[s] = wmma_f32(a, bI, aI[s]);
      }
    }
  }
  int lane = threadIdx.x & 31, n = lane & 15, hi = lane >> 4;
  size_t base = (size_t)img * IMSZ;
#pragma unroll
  for (int s = 0; s < 4; ++s)
#pragma unroll
    for (int r = 0; r < 8; ++r) {
      int gy = r0 + (w << 4) + (hi << 3) + r;
      int gx = c0 + (s << 4) + n;
      size_t idx = base + (size_t)gy * NN + gx;
      outR[idx] = aR[s][r];
      outI[idx] = sI * aI[s][r];
    }
}

// ---- pass B: P = (C@Yr + S@Yi)^2 + (C@Yi - S@Yr)^2  (col DFT + power) -----
__global__ __launch_bounds__(128) void ck_coldft_pow(const float* __restrict__ Yr,
                                                     const float* __restrict__ Yi,
                                                     const float* __restrict__ Cm,
                                                     const float* __restrict__ Sm,
                                                     float* __restrict__ P) {
  __shared__ float Ac[64 * LDA];
  __shared__ float Av[64 * LDA];
  __shared__ float Br[16 * LDBP];
  __shared__ float Bi[16 * LDBP];
  int img = blockIdx.x >> 4;
  int tl  = blockIdx.x & 15;
  int r0 = (tl >> 2) << 6, c0 = (tl & 3) << 6;
  size_t base = (size_t)img * IMSZ;
  int w = threadIdx.x >> 5;

  v8f zr[4], zi[4];
#pragma unroll
  for (int s = 0; s < 4; ++s) { zr[s] = vzero8(); zi[s] = vzero8(); }

  for (int kc = 0; kc < 8; ++kc) {
    __syncthreads();
    load_tile_A(Ac, Cm + (size_t)r0 * NN + kc * 32, NN);
    load_tile_A(Av, Sm + (size_t)r0 * NN + kc * 32, NN);
    load_tile_B(Br, Yr + base + (size_t)(kc * 32) * NN + c0, NN);
    load_tile_B(Bi, Yi + base + (size_t)(kc * 32) * NN + c0, NN);
    __syncthreads();
    const float* Acw = Ac + (w << 4) * LDA;
    const float* Asw = Av + (w << 4) * LDA;
#pragma unroll
    for (int ks = 0; ks < 8; ++ks) {
      v2f aC = fragA(Acw, ks);
      v2f aS = fragA(Asw, ks);
      v2f aSn; aSn.x = -aS.x; aSn.y = -aS.y;   // f32 WMMA has no A-neg modifier
#pragma unroll
      for (int s = 0; s < 4; ++s) {
        v2f bR = fragB(Br, ks, s << 4);
        v2f bI = fragB(Bi, ks, s << 4);
        zr[s] = wmma_f32(aC, bR, zr[s]);
        zr[s] = wmma_f32(aS, bI, zr[s]);
        zi[s] = wmma_f32(aC, bI, zi[s]);
        zi[s] = wmma_f32(aSn, bR, zi[s]);
      }
    }
  }
  int lane = threadIdx.x & 31, n = lane & 15, hi = lane >> 4;
#pragma unroll
  for (int s = 0; s < 4; ++s)
#pragma unroll
    for (int r = 0; r < 8; ++r) {
      int gy = r0 + (w << 4) + (hi << 3) + r;
      int gx = c0 + (s << 4) + n;
      float a = zr[s][r], b = zi[s][r];
      P[base + (size_t)gy * NN + gx] = a * a + b * b;
    }
}

// ---- pass C2: out = SCALE*(C@T1 - S@T2), scattered back to NHWC -----------
__global__ __launch_bounds__(128) void ck_inv_scatter(const float* __restrict__ T1,
                                                      const float* __restrict__ T2,
                                                      const float* __restrict__ Cm,
                                                      const float* __restrict__ Sm,
                                                      float* __restrict__ out) {
  __shared__ float Ac[64 * LDA];
  __shared__ float Av[64 * LDA];
  __shared__ float B1[16 * LDBP];
  __shared__ float B2[16 * LDBP];
  int img = blockIdx.x >> 4;
  int tl  = blockIdx.x & 15;
  int r0 = (tl >> 2) << 6, c0 = (tl & 3) << 6;
  size_t base = (size_t)img * IMSZ;
  int w = threadIdx.x >> 5;

  v8f z[4];
#pragma unroll
  for (int s = 0; s < 4; ++s) z[s] = vzero8();

  for (int kc = 0; kc < 8; ++kc) {
    __syncthreads();
    load_tile_A(Ac, Cm + (size_t)r0 * NN + kc * 32, NN);
    load_tile_A(Av, Sm + (size_t)r0 * NN + kc * 32, NN);
    load_tile_B(B1, T1 + base + (size_t)(kc * 32) * NN + c0, NN);
    load_tile_B(B2, T2 + base + (size_t)(kc * 32) * NN + c0, NN);
    __syncthreads();
    const float* Acw = Ac + (w << 4) * LDA;
    const float* Asw = Av + (w << 4) * LDA;
#pragma unroll
    for (int ks = 0; ks < 8; ++ks) {
      v2f aC = fragA(Acw, ks);
      v2f aS = fragA(Asw, ks);
      v2f aSn; aSn.x = -aS.x; aSn.y = -aS.y;
#pragma unroll
      for (int s = 0; s < 4; ++s) {
        v2f b1 = fragB(B1, ks, s << 4);
        v2f b2 = fragB(B2, ks, s << 4);
        z[s] = wmma_f32(aC, b1, z[s]);
        z[s] = wmma_f32(aSn, b2, z[s]);
      }
    }
  }
  int lane = threadIdx.x & 31, n = lane & 15, hi = lane >> 4;
  int b = img >> 3, c = img & 7;
#pragma unroll
  for (int s = 0; s < 4; ++s)
#pragma unroll
    for (int r = 0; r < 8; ++r) {
      int gy = r0 + (w << 4) + (hi << 3) + r;
      int gx = c0 + (s << 4) + n;
      out[(((size_t)b * NN + gy) * NN + gx) * 8 + c] = z[s][r] * SCALE;
    }
}

// ---------------------------------------------------------------------------
extern "C" void kernel_launch(void* const* d_in, const int* in_sizes, int n_in,
                              void* d_out, int out_size, void* d_ws, size_t ws_size,
                              hipStream_t stream) {
  const float* in = (const float*)d_in[0];
  float* out = (float*)d_out;
  float* ws  = (float*)d_ws;

  const size_t IMGS = (size_t)NIMG * IMSZ;      // 33,554,432 floats per buffer
  float* buf0 = ws;                             // d, later P
  float* buf1 = ws + IMGS;                      // Yr, later T1 = P@C
  float* buf2 = ws + 2 * IMGS;                  // Yi, later T2 = P@S
  float* Cm   = ws + 3 * IMGS;                  // 65536 floats
  float* Sm   = Cm + IMSZ;                      // 65536 floats
  float* part = Sm + IMSZ;                      // 4096 floats (mean partials)

  ck_twiddles<<<256, 256, 0, stream>>>(Cm, Sm);
  ck_partial<<<512, 256, 0, stream>>>(in, part);
  ck_center<<<512, 256, 0, stream>>>(in, part, buf0);
  // Row DFT: Yr = d@C, Yi = -d@S
  ck_rowdft<<<NIMG * 16, 128, 0, stream>>>(buf0, Cm, Sm, buf1, buf2, -1.0f);
  // Column DFT + power: P = Zr^2 + Zi^2  (writes over d)
  ck_coldft_pow<<<NIMG * 16, 128, 0, stream>>>(buf1, buf2, Cm, Sm, buf0);
  // Inverse, stage 1 (reuse row-DFT kernel): T1 = P@C, T2 = P@S
  ck_rowdft<<<NIMG * 16, 128, 0, stream>>>(buf0, Cm, Sm, buf1, buf2, 1.0f);
  // Inverse, stage 2 + scale + NHWC scatter
  ck_inv_scatter<<<NIMG * 16, 128, 0, stream>>>(buf1, buf2, Cm, Sm, out);
}